// DifferentiableBiquadChain_20675972563312
// MI455X (gfx1250) — compile-verified
//
#include <hip/hip_runtime.h>
#include <math.h>

typedef __attribute__((ext_vector_type(2))) float v2f;
typedef __attribute__((ext_vector_type(8))) float v8f;

#define NBQ    16
#define FRAME  2048
#define NF     128
#define NBATCH 16
#define NCH    (NBATCH*NF)   // 2048 channels (one wave each)
#define WPW    4             // waves (channels) per workgroup
#define LBLK   16            // samples per block
#define MBLK   16            // blocks per macroblock (256 samples)
#define NMB    (FRAME/(LBLK*MBLK)) // 8 macroblocks

// All matrix work via V_WMMA_F32_16X16X4_F32 (f32, wave32).
__launch_bounds__(WPW*32)
__global__ void eq_chain_kernel(const float* __restrict__ audio,
                                const float* __restrict__ params,
                                float* __restrict__ out)
{
    __shared__ float sh_coef[WPW][NBQ*5];
    __shared__ float sh_Theta[WPW][16*32];  // y-from-state map over one block
    __shared__ float sh_A16[WPW][32*32];    // A^16 (block state transition)
    __shared__ float sh_Gam[WPW][32*16];    // input->state map over one block
    __shared__ float sh_h[WPW][16];         // cascade impulse response
    __shared__ float sh_S[WPW][32*16];      // block-start states of a macroblock
    __shared__ float sh_G[WPW][32*16];      // WMMA Gamma*X dump
    __shared__ float sh_s[WPW][32];         // running state exchange

    const int lane = threadIdx.x & 31;
    const int wv   = threadIdx.x >> 5;
    const int ch   = blockIdx.x * WPW + wv;
    const int bb   = ch >> 7;
    const int ff   = ch & (NF-1);

    float* coef  = sh_coef[wv];
    float* Theta = sh_Theta[wv];
    float* A16   = sh_A16[wv];
    float* Gam   = sh_Gam[wv];
    float* hbuf  = sh_h[wv];
    float* Smat  = sh_S[wv];
    float* Gbuf  = sh_G[wv];
    float* sbuf  = sh_s[wv];

    // ---- broadband gains (all lanes, broadcast loads) ----
    const float* prm = params + bb*50*NF + ff;
    float ing_db  = -60.f + 60.f * prm[48*NF];
    float outg_db = -60.f + 60.f * prm[49*NF];
    float in_g  = __expf(ing_db  * 0.115129254649702f);  // ln(10)/20
    float out_g = __expf(outg_db * 0.115129254649702f);

    // ---- per-stage biquad coefficients (lane i -> stage i) ----
    if (lane < NBQ) {
        int i = lane;
        float fn = prm[(3*i+0)*NF];
        float gn = prm[(3*i+1)*NF];
        float qn = prm[(3*i+2)*NF];
        const float lq0 = -0.6931471805599453f;  // ln 0.5
        const float lq1 =  2.772588722239781f;   // ln 16
        float Q = __expf(lq0 + qn*(lq1-lq0));
        float flo, fhi;
        if (i==0)                   { flo=20.f;   fhi=500.f;   }
        else if (i==NBQ-1)          { flo=5000.f; fhi=20000.f; }
        else if (i==1 || i==NBQ-2)  { flo=50.f;   fhi=16000.f; }
        else                        { flo=100.f;  fhi=15000.f; }
        float llo = __logf(flo), lhi = __logf(fhi);
        float fc = __expf(llo + fn*(lhi-llo));
        float w0 = 6.283185307179586f * fc / 96000.f;
        float sn, cs;
        __sincosf(w0, &sn, &cs);
        float al = sn / (2.f * Q);
        float b0,b1,b2,a0,a1,a2;
        if (i==0) {                 // highpass
            b0 = (1.f+cs)*0.5f; b1 = -(1.f+cs); b2 = b0;
            a0 = 1.f+al; a1 = -2.f*cs; a2 = 1.f-al;
        } else if (i==NBQ-1) {      // lowpass
            b0 = (1.f-cs)*0.5f; b1 = 1.f-cs; b2 = b0;
            a0 = 1.f+al; a1 = -2.f*cs; a2 = 1.f-al;
        } else {
            float gdb = -24.f + gn*48.f;
            float A = __expf(gdb * 0.05756462732485114f); // ln(10)/40
            if (i==1) {             // low shelf
                float sA = sqrtf(A);
                b0 = A*((A+1.f) - (A-1.f)*cs + 2.f*sA*al);
                b1 = 2.f*A*((A-1.f) - (A+1.f)*cs);
                b2 = A*((A+1.f) - (A-1.f)*cs - 2.f*sA*al);
                a0 = (A+1.f) + (A-1.f)*cs + 2.f*sA*al;
                a1 = -2.f*((A-1.f) + (A+1.f)*cs);
                a2 = (A+1.f) + (A-1.f)*cs - 2.f*sA*al;
            } else if (i==NBQ-2) {  // high shelf
                float sA = sqrtf(A);
                b0 = A*((A+1.f) + (A-1.f)*cs + 2.f*sA*al);
                b1 = -2.f*A*((A-1.f) + (A+1.f)*cs);
                b2 = A*((A+1.f) + (A-1.f)*cs - 2.f*sA*al);
                a0 = (A+1.f) - (A-1.f)*cs + 2.f*sA*al;
                a1 = 2.f*((A-1.f) - (A+1.f)*cs);
                a2 = (A+1.f) - (A-1.f)*cs - 2.f*sA*al;
            } else {                // peaking
                b0 = 1.f + al*A; b1 = -2.f*cs; b2 = 1.f - al*A;
                a0 = 1.f + al/A; a1 = -2.f*cs; a2 = 1.f - al/A;
            }
        }
        float inv = 1.f/a0;
        coef[i*5+0] = b0*inv;
        coef[i*5+1] = b1*inv;
        coef[i*5+2] = b2*inv;
        coef[i*5+3] = a1*inv;
        coef[i*5+4] = a2*inv;
    }
    __syncthreads();

    float cb0[NBQ], cb1[NBQ], cb2[NBQ], ca1[NBQ], ca2[NBQ];
    #pragma unroll
    for (int i=0;i<NBQ;i++){
        cb0[i]=coef[i*5+0]; cb1[i]=coef[i*5+1]; cb2[i]=coef[i*5+2];
        ca1[i]=coef[i*5+3]; ca2[i]=coef[i*5+4];
    }

    // ---- unit-state simulations: lane l starts with unit in state l ----
    // gives Theta[t][l] (outputs) and A^16 column l (final state)
    float s1[NBQ], s2[NBQ];
    #pragma unroll
    for (int i=0;i<NBQ;i++){
        s1[i] = (lane == 2*i  ) ? 1.f : 0.f;
        s2[i] = (lane == 2*i+1) ? 1.f : 0.f;
    }
    for (int t=0;t<LBLK;t++){
        float v = 0.f;
        #pragma unroll
        for (int i=0;i<NBQ;i++){
            float y   = fmaf(cb0[i], v, s1[i]);
            float ns1 = fmaf(cb1[i], v, fmaf(-ca1[i], y, s2[i]));
            s2[i] = fmaf(cb2[i], v, -ca2[i]*y);
            s1[i] = ns1;
            v = y;
        }
        Theta[t*32 + lane] = v;
    }
    #pragma unroll
    for (int i=0;i<NBQ;i++){
        A16[(2*i  )*32 + lane] = s1[i];
        A16[(2*i+1)*32 + lane] = s2[i];
    }

    // ---- impulse simulation: gives h[t] (Phi Toeplitz) and Gamma columns ----
    #pragma unroll
    for (int i=0;i<NBQ;i++){ s1[i]=0.f; s2[i]=0.f; }
    for (int t=0;t<LBLK;t++){
        float v = (t==0) ? 1.f : 0.f;
        #pragma unroll
        for (int i=0;i<NBQ;i++){
            float y   = fmaf(cb0[i], v, s1[i]);
            float ns1 = fmaf(cb1[i], v, fmaf(-ca1[i], y, s2[i]));
            s2[i] = fmaf(cb2[i], v, -ca2[i]*y);
            s1[i] = ns1;
            v = y;
        }
        if (lane == 0) hbuf[t] = v;
        // Gamma[:, 15-t] = A^t * b = state after (t+1) steps
        float sel = 0.f;
        #pragma unroll
        for (int i=0;i<NBQ;i++){
            sel = (lane == 2*i  ) ? s1[i] : sel;
            sel = (lane == 2*i+1) ? s2[i] : sel;
        }
        Gam[lane*16 + (15-t)] = sel;
    }
    __syncthreads();

    // ---- WMMA A-operand fragments ----
    const int Mrow = lane & 15;
    const int kb   = (lane >> 4) * 2;  // lanes 16-31 hold K+2 / K+3
    v2f fPhi[4], fGt[4], fGb[4];
    #pragma unroll
    for (int q=0;q<4;q++){
        int k0 = 4*q + kb;
        int k1 = k0 + 1;
        fPhi[q].x = (Mrow >= k0) ? hbuf[Mrow-k0] : 0.f;
        fPhi[q].y = (Mrow >= k1) ? hbuf[Mrow-k1] : 0.f;
        fGt[q].x = Gam[Mrow*16 + k0];
        fGt[q].y = Gam[Mrow*16 + k1];
        fGb[q].x = Gam[(16+Mrow)*16 + k0];
        fGb[q].y = Gam[(16+Mrow)*16 + k1];
    }
    // Theta as A-operand (16x32 -> 8 K-chunks)
    v2f fTh[8];
    #pragma unroll
    for (int q=0;q<8;q++){
        int k0 = 4*q + kb;
        fTh[q].x = Theta[Mrow*32 + k0];
        fTh[q].y = Theta[Mrow*32 + k0 + 1];
    }
    // A^16 row for the serial state recursion (lane = row)
    float Ar[32];
    #pragma unroll
    for (int k=0;k<32;k++) Ar[k] = A16[lane*32 + k];

    const float* xin = audio + ch*FRAME;
    float* yo = out + ch*FRAME;

    sbuf[lane] = 0.f;
    float myS = 0.f;   // this lane's element of the running state

    for (int m=0;m<NMB;m++){
        if (m+1 < NMB) __builtin_prefetch(xin + (m+1)*256 + lane*8, 0, 1);
        // B-operand: X[k][j] = x[m*256 + j*16 + k] * in_g (column j = block j)
        v2f fX[4];
        #pragma unroll
        for (int q=0;q<4;q++){
            int k0 = 4*q + kb;
            fX[q].x = in_g * xin[m*256 + Mrow*16 + k0];
            fX[q].y = in_g * xin[m*256 + Mrow*16 + k0 + 1];
        }
        v8f dY  = {0.f,0.f,0.f,0.f,0.f,0.f,0.f,0.f};
        v8f dGt = dY, dGb = dY;
        #pragma unroll
        for (int q=0;q<4;q++){
            dY  = __builtin_amdgcn_wmma_f32_16x16x4_f32(false, fPhi[q], false, fX[q], (short)0, dY,  false, false);
            dGt = __builtin_amdgcn_wmma_f32_16x16x4_f32(false, fGt[q],  false, fX[q], (short)0, dGt, false, false);
            dGb = __builtin_amdgcn_wmma_f32_16x16x4_f32(false, fGb[q],  false, fX[q], (short)0, dGb, false, false);
        }
        // dump Gamma*X (rows 0..31 x 16 block-columns) for the serial loop
        const int hi = (lane >> 4) * 8;
        #pragma unroll
        for (int r=0;r<8;r++){
            Gbuf[(r+hi)*16 + Mrow]    = dGt[r];
            Gbuf[(16+r+hi)*16 + Mrow] = dGb[r];
        }
        // serial block recursion: s_{j+1} = A16*s_j + G[:,j]; record S[:,j]=s_j
        for (int j=0;j<MBLK;j++){
            Smat[lane*16 + j] = myS;       // column j = state at block start
            float sv[32];
            #pragma unroll
            for (int k=0;k<32;k++) sv[k] = sbuf[k];
            float p0=0.f, p1=0.f, p2=0.f, p3=0.f;
            #pragma unroll
            for (int k=0;k<8;k++){
                p0 = fmaf(Ar[4*k+0], sv[4*k+0], p0);
                p1 = fmaf(Ar[4*k+1], sv[4*k+1], p1);
                p2 = fmaf(Ar[4*k+2], sv[4*k+2], p2);
                p3 = fmaf(Ar[4*k+3], sv[4*k+3], p3);
            }
            float acc = Gbuf[lane*16 + j] + ((p0+p1)+(p2+p3));
            sbuf[lane] = acc;   // all lanes read sv before this store (lockstep wave)
            myS = acc;
        }
        // Ycorr = Theta(16x32) * S(32x16) via 8 WMMAs, then Y = Yzs + Ycorr
        v8f dYc = {0.f,0.f,0.f,0.f,0.f,0.f,0.f,0.f};
        #pragma unroll
        for (int q=0;q<8;q++){
            int k0 = 4*q + kb;
            v2f fS;
            fS.x = Smat[ k0      *16 + Mrow];
            fS.y = Smat[(k0 + 1) *16 + Mrow];
            dYc = __builtin_amdgcn_wmma_f32_16x16x4_f32(false, fTh[q], false, fS, (short)0, dYc, false, false);
        }
        v8f yf = (dY + dYc) * out_g;
        // D layout: element r -> sample t = r+hi of block j = Mrow
        float* dst = yo + m*256 + Mrow*16 + hi;
        float4 lo4; lo4.x=yf[0]; lo4.y=yf[1]; lo4.z=yf[2]; lo4.w=yf[3];
        float4 hi4; hi4.x=yf[4]; hi4.y=yf[5]; hi4.z=yf[6]; hi4.w=yf[7];
        *(float4*)(dst)     = lo4;
        *(float4*)(dst + 4) = hi4;
    }
}

extern "C" void kernel_launch(void* const* d_in, const int* in_sizes, int n_in,
                              void* d_out, int out_size, void* d_ws, size_t ws_size,
                              hipStream_t stream) {
    (void)in_sizes; (void)n_in; (void)out_size; (void)d_ws; (void)ws_size;
    const float* audio  = (const float*)d_in[0];
    const float* params = (const float*)d_in[1];
    float* out = (float*)d_out;
    dim3 grid(NCH / WPW);
    dim3 block(WPW * 32);
    hipLaunchKernelGGL(eq_chain_kernel, grid, block, 0, stream, audio, params, out);
}